// CrossEntropyMoreToMore_60413009986132
// MI455X (gfx1250) — compile-verified
//
#include <hip/hip_runtime.h>
#include <hip/hip_bf16.h>

typedef __attribute__((ext_vector_type(2))) float v2f;
typedef __attribute__((ext_vector_type(4))) float f4;
typedef __attribute__((ext_vector_type(8))) float v8f;

__device__ __forceinline__ float softplus_f(float x) {
    // Numerically stable: max(x,0) + log1p(exp(-|x|))  (== jax.nn.softplus)
    return fmaxf(x, 0.0f) + log1pf(expf(-fabsf(x)));
}

// ---------------------------------------------------------------------------
// Phase 1: base[n] = sum_c softplus(pre_cls[n, c])
// One wave handles 16 rows (n0..n0+15) via V_WMMA_F32_16X16X4_F32 with B=ones.
// A-matrix (16x4 f32) lane layout: lanes 0-15 -> M=lane, K={0,1}; lanes 16-31
// -> M=lane-16, K={2,3}. B = all ones, so its lane layout is irrelevant.
// D (16x16 f32, 8 VGPRs): vgpr r, lanes 0-15 -> M=r; lanes 16-31 -> M=r+8.
// Tail handling (C%4 != 0) is branchless: clamped in-bounds loads + cndmask,
// so control flow stays uniform and EXEC stays all-1s for every WMMA.
// ---------------------------------------------------------------------------
__global__ void cemm_base_wmma(const float* __restrict__ pre,
                               float* __restrict__ base, int N, int C) {
    const int lane = threadIdx.x & 31;
    const int wave = (blockIdx.x * (blockDim.x >> 5)) + (threadIdx.x >> 5);
    const int n0   = wave * 16;
    if (n0 >= N) return;  // wave-uniform exit (n0 uniform within a wave)

    const int mrow = lane & 15;
    const int kb   = (lane >> 4) << 1;  // 0 for lanes 0-15, 2 for lanes 16-31
    const float* rowbase = pre + (size_t)(n0 + mrow) * C;

    v8f acc = {};
    v2f ones; ones.x = 1.0f; ones.y = 1.0f;

    for (int c0 = 0; c0 < C; c0 += 4) {
        const int ca = c0 + kb;
        const int cb = ca + 1;
        // Clamped (always in-bounds) loads: no divergent branches, no EXEC ops.
        float x0 = rowbase[min(ca, C - 1)];
        float x1 = rowbase[min(cb, C - 1)];
        v2f a;
        a.x = (ca < C) ? softplus_f(x0) : 0.0f;  // v_cndmask, not a branch
        a.y = (cb < C) ? softplus_f(x1) : 0.0f;
        // D = A x ones(4x16) + C  -> every column of D holds the row partial sum
        acc = __builtin_amdgcn_wmma_f32_16x16x4_f32(
            /*neg_a=*/false, a, /*neg_b=*/false, ones,
            /*c_mod=*/(short)0, acc, /*reuse_a=*/false, /*reuse_b=*/false);
    }

    // Extract row sums: lane 0 holds rows n0..n0+7 in acc[0..7],
    // lane 16 holds rows n0+8..n0+15 in acc[0..7].
    if (lane == 0) {
        #pragma unroll
        for (int r = 0; r < 8; ++r) base[n0 + r] = acc[r];
    } else if (lane == 16) {
        #pragma unroll
        for (int r = 0; r < 8; ++r) base[n0 + 8 + r] = acc[r];
    }
}

// ---------------------------------------------------------------------------
// Phase 2: LUT rows[c][n] = base[n] - pre_cls[n, c]   (80 x 8192 = 2.56 MB,
// stays resident in the 192 MB L2 for phase 3)
// ---------------------------------------------------------------------------
__global__ void cemm_build_lut(const float* __restrict__ base,
                               const float* __restrict__ pre,
                               float* __restrict__ rows, int N, int C) {
    int n = blockIdx.x * blockDim.x + threadIdx.x;
    if (n >= N) return;
    float b = base[n];
    const float* p = pre + (size_t)n * C;
    for (int c = 0; c < C; ++c)
        rows[(size_t)c * N + n] = b - p[c];
}

// ---------------------------------------------------------------------------
// Phase 3: out[m, :] = rows[idx[m]]  — the bandwidth-bound part.
// Contiguous B128 reads (L2-resident LUT) -> non-temporal B128 stores (HBM).
// One block per output row, 256 threads, float4 per thread per iteration.
// ---------------------------------------------------------------------------
__global__ void cemm_broadcast_lut(const float* __restrict__ rows,
                                   const int* __restrict__ idx,
                                   float* __restrict__ out, int N) {
    const int m = blockIdx.x;
    const int c = idx[m];  // uniform per block -> scalar load
    const f4* __restrict__ src = (const f4*)(rows + (size_t)c * N);
    f4* __restrict__ dst = (f4*)(out + (size_t)m * N);
    const int n4 = N >> 2;
    __builtin_prefetch(src + threadIdx.x, 0, 1);  // global_prefetch_b8 into L2
    #pragma unroll 4
    for (int j = threadIdx.x; j < n4; j += blockDim.x) {
        f4 v = src[j];
        __builtin_nontemporal_store(v, &dst[j]);  // streaming store, TH=NT
    }
}

// Fallback (tiny workspace): out[m,n] = base[n] - pre_cls[n, idx[m]] directly.
__global__ void cemm_broadcast_direct(const float* __restrict__ base,
                                      const float* __restrict__ pre,
                                      const int* __restrict__ idx,
                                      float* __restrict__ out, int N, int C) {
    const int m = blockIdx.x;
    const int c = idx[m];
    const f4* __restrict__ b4 = (const f4*)base;
    f4* __restrict__ dst = (f4*)(out + (size_t)m * N);
    const int n4 = N >> 2;
    for (int j = threadIdx.x; j < n4; j += blockDim.x) {
        f4 b = b4[j];
        int n = j << 2;
        f4 v;
        v.x = b.x - pre[(size_t)(n + 0) * C + c];
        v.y = b.y - pre[(size_t)(n + 1) * C + c];
        v.z = b.z - pre[(size_t)(n + 2) * C + c];
        v.w = b.w - pre[(size_t)(n + 3) * C + c];
        __builtin_nontemporal_store(v, &dst[j]);
    }
}

extern "C" void kernel_launch(void* const* d_in, const int* in_sizes, int n_in,
                              void* d_out, int out_size, void* d_ws, size_t ws_size,
                              hipStream_t stream) {
    const int*   idx = (const int*)d_in[0];    // gt_kind_ind, (M,) int32
    const float* pre = (const float*)d_in[1];  // pre_cls, (N, C) f32
    float* out = (float*)d_out;                // (M, N) f32

    const int M = in_sizes[0];
    const int N = out_size / M;
    const int C = in_sizes[1] / N;

    float* base = (float*)d_ws;                           // N floats
    size_t base_bytes = ((size_t)N * sizeof(float) + 255) & ~(size_t)255;
    float* rows = (float*)((char*)d_ws + base_bytes);     // C*N floats
    size_t need_lut = base_bytes + (size_t)C * N * sizeof(float);

    // Phase 1: base[] via WMMA row-reduction. 16 rows per wave, 8 waves/block.
    {
        int waves  = (N + 15) / 16;
        int blocks = (waves + 7) / 8;
        cemm_base_wmma<<<blocks, 256, 0, stream>>>(pre, base, N, C);
    }

    if (ws_size >= need_lut) {
        // Phase 2: 80-row LUT (L2-resident).
        cemm_build_lut<<<(N + 255) / 256, 256, 0, stream>>>(base, pre, rows, N, C);
        // Phase 3: broadcast rows -> out with NT streaming stores.
        cemm_broadcast_lut<<<M, 256, 0, stream>>>(rows, idx, out, N);
    } else {
        cemm_broadcast_direct<<<M, 256, 0, stream>>>(base, pre, idx, out, N, C);
    }
}